// Mixhop_Net_26439818674272
// MI455X (gfx1250) — compile-verified
//
#include <hip/hip_runtime.h>
#include <hip/hip_bf16.h>
#include <math.h>

typedef __attribute__((ext_vector_type(2))) float v2f;
typedef __attribute__((ext_vector_type(8))) float v8f;

#define F_IN 256
#define HDIM 64
#define CDIM 40
#define K3   192  // 3*HDIM
#define CP   48   // CDIM padded to 3 WMMA col-tiles

// ---------------- degree / GCN normalization ----------------
__global__ void k_deg_init(float* __restrict__ deg, int n) {
  int i = blockIdx.x * blockDim.x + threadIdx.x;
  if (i < n) deg[i] = 1.0f;  // self loop contributes 1 to every node's degree
}

__global__ void k_deg_count(const int* __restrict__ dst, float* __restrict__ deg, int e) {
  int i = blockIdx.x * blockDim.x + threadIdx.x;
  if (i < e) atomicAdd(&deg[dst[i]], 1.0f);
}

__global__ void k_dis(const float* __restrict__ deg, float* __restrict__ dis, int n) {
  int i = blockIdx.x * blockDim.x + threadIdx.x;
  if (i < n) {
    float d = deg[i];
    dis[i] = (d > 0.0f) ? rsqrtf(d) : 0.0f;
  }
}

// per-edge norm = dis[src] * dis[dst]  (read once instead of 64x in the scatter)
__global__ void k_norm(const int* __restrict__ src, const int* __restrict__ dst,
                       const float* __restrict__ dis, float* __restrict__ norm, int e) {
  int i = blockIdx.x * blockDim.x + threadIdx.x;
  if (i < e) norm[i] = dis[src[i]] * dis[dst[i]];
}

// pad w2 [192,40] -> [192,48] and b2 [40] -> [48] with zeros
__global__ void k_padw2(const float* __restrict__ w2, const float* __restrict__ b2,
                        float* __restrict__ w2p, float* __restrict__ b2p) {
  int i = blockIdx.x * blockDim.x + threadIdx.x;
  if (i < K3 * CP) {
    int r = i / CP, c = i % CP;
    w2p[i] = (c < CDIM) ? w2[r * CDIM + c] : 0.0f;
  } else if (i < K3 * CP + CP) {
    int c = i - K3 * CP;
    b2p[c] = (c < CDIM) ? b2[c] : 0.0f;
  }
}

// ---------------- WMMA f32 GEMMs ----------------
// h = relu(X[N,256] @ W[256,64] + b).  8 waves/block: 2 row-tiles x 4 col-tiles.
__global__ void k_lin1(const float* __restrict__ X, const float* __restrict__ W,
                       const float* __restrict__ bias, float* __restrict__ Hout, int n) {
  const int wave = threadIdx.x >> 5;
  const int lane = threadIdx.x & 31;
  const int half = lane >> 4;      // 0: K pair {0,1}, 1: K pair {2,3}
  const int m    = lane & 15;
  const int rowBase = blockIdx.x * 32 + (wave >> 2) * 16;
  const int colBase = (wave & 3) * 16;
  const int col = colBase + m;
  int arow = rowBase + m; if (arow > n - 1) arow = n - 1;
  const float* __restrict__ xr = X + (size_t)arow * F_IN;
  v8f c = {};
  #pragma unroll 8
  for (int k0 = 0; k0 < F_IN; k0 += 4) {
    const int ka = k0 + 2 * half;
    v2f a; a.x = xr[ka]; a.y = xr[ka + 1];
    v2f b; b.x = W[(size_t)ka * HDIM + col]; b.y = W[(size_t)(ka + 1) * HDIM + col];
    c = __builtin_amdgcn_wmma_f32_16x16x4_f32(false, a, false, b, (short)0, c, false, false);
  }
  const float bv = bias[col];
  float* __restrict__ op = Hout + (size_t)(rowBase + 8 * half) * HDIM + col;
  if (rowBase + 16 <= n) {           // uniform fast path: 8 unconditional stores
    #pragma unroll
    for (int r = 0; r < 8; ++r) {
      float v = c[r] + bv;
      op[(size_t)r * HDIM] = v > 0.0f ? v : 0.0f;
    }
  } else {
    #pragma unroll
    for (int r = 0; r < 8; ++r) {
      const int row = rowBase + r + 8 * half;
      float v = c[r] + bv;
      if (row < n) op[(size_t)r * HDIM] = v > 0.0f ? v : 0.0f;
    }
  }
}

// O[:, block] = relu(V[N,64] @ W[64,64] + b), O has row stride 192 (pre-offset by column block)
__global__ void k_linp(const float* __restrict__ V, const float* __restrict__ W,
                       const float* __restrict__ bias, float* __restrict__ O, int n) {
  const int wave = threadIdx.x >> 5;
  const int lane = threadIdx.x & 31;
  const int half = lane >> 4;
  const int m    = lane & 15;
  const int rowBase = blockIdx.x * 32 + (wave >> 2) * 16;
  const int colBase = (wave & 3) * 16;
  const int col = colBase + m;
  int arow = rowBase + m; if (arow > n - 1) arow = n - 1;
  const float* __restrict__ vr = V + (size_t)arow * HDIM;
  v8f c = {};
  #pragma unroll
  for (int k0 = 0; k0 < HDIM; k0 += 4) {
    const int ka = k0 + 2 * half;
    v2f a; a.x = vr[ka]; a.y = vr[ka + 1];
    v2f b; b.x = W[(size_t)ka * HDIM + col]; b.y = W[(size_t)(ka + 1) * HDIM + col];
    c = __builtin_amdgcn_wmma_f32_16x16x4_f32(false, a, false, b, (short)0, c, false, false);
  }
  const float bv = bias[col];
  float* __restrict__ op = O + (size_t)(rowBase + 8 * half) * K3 + col;
  if (rowBase + 16 <= n) {
    #pragma unroll
    for (int r = 0; r < 8; ++r) {
      float v = c[r] + bv;
      op[(size_t)r * K3] = v > 0.0f ? v : 0.0f;  // relu(concat) == relu per part
    }
  } else {
    #pragma unroll
    for (int r = 0; r < 8; ++r) {
      const int row = rowBase + r + 8 * half;
      float v = c[r] + bv;
      if (row < n) op[(size_t)r * K3] = v > 0.0f ? v : 0.0f;
    }
  }
}

// logits[N,48] = V[N,192] @ W2p[192,48] + b2p.  6 waves/block: 2 row x 3 col tiles.
__global__ void k_lin2(const float* __restrict__ V, const float* __restrict__ W,
                       const float* __restrict__ bias, float* __restrict__ L, int n) {
  const int wave = threadIdx.x >> 5;
  const int lane = threadIdx.x & 31;
  const int half = lane >> 4;
  const int m    = lane & 15;
  const int rowBase = blockIdx.x * 32 + (wave / 3) * 16;
  const int colBase = (wave % 3) * 16;
  const int col = colBase + m;
  int arow = rowBase + m; if (arow > n - 1) arow = n - 1;
  const float* __restrict__ vr = V + (size_t)arow * K3;
  v8f c = {};
  #pragma unroll 8
  for (int k0 = 0; k0 < K3; k0 += 4) {
    const int ka = k0 + 2 * half;
    v2f a; a.x = vr[ka]; a.y = vr[ka + 1];
    v2f b; b.x = W[(size_t)ka * CP + col]; b.y = W[(size_t)(ka + 1) * CP + col];
    c = __builtin_amdgcn_wmma_f32_16x16x4_f32(false, a, false, b, (short)0, c, false, false);
  }
  const float bv = bias[col];
  float* __restrict__ op = L + (size_t)(rowBase + 8 * half) * CP + col;
  if (rowBase + 16 <= n) {
    #pragma unroll
    for (int r = 0; r < 8; ++r) op[(size_t)r * CP] = c[r] + bv;
  } else {
    #pragma unroll
    for (int r = 0; r < 8; ++r) {
      const int row = rowBase + r + 8 * half;
      if (row < n) op[(size_t)r * CP] = c[r] + bv;
    }
  }
}

// ---------------- propagation: out = D^-1/2 (A + I) D^-1/2 v ----------------
// self-loop term as initialization: out[i] = dis[i]^2 * v[i]
__global__ void k_prop_init(const float* __restrict__ V, const float* __restrict__ dis,
                            float* __restrict__ O, int n) {
  int i = blockIdx.x * blockDim.x + threadIdx.x;  // n*16 threads, float4 each
  int node = i >> 4;
  if (node >= n) return;
  int f = (i & 15) * 4;
  float d = dis[node];
  float s = d * d;
  const float4 val = *(const float4*)(V + (size_t)node * HDIM + f);
  float4 o; o.x = s * val.x; o.y = s * val.y; o.z = s * val.z; o.w = s * val.w;
  *(float4*)(O + (size_t)node * HDIM + f) = o;
}

// edge scatter: out[dst] += norm[e] * v[src]; 32 threads per edge, float2 per thread
__global__ void k_prop_edge(const int* __restrict__ src, const int* __restrict__ dst,
                            const float* __restrict__ norm, const float* __restrict__ V,
                            float* __restrict__ O, int e) {
  unsigned long long t = (unsigned long long)blockIdx.x * blockDim.x + threadIdx.x;
  int ed = (int)(t >> 5);
  if (ed >= e) return;
  int f = (int)(t & 31) * 2;
  int s = src[ed], d = dst[ed];
  float nrm = norm[ed];
  const float2 v = *(const float2*)(V + (size_t)s * HDIM + f);
  float* o = O + (size_t)d * HDIM + f;
  atomicAdd(o,     nrm * v.x);
  atomicAdd(o + 1, nrm * v.y);
}

// ---------------- log_softmax over C=40 (reads padded stride 48) ----------------
__global__ void k_logsoftmax(const float* __restrict__ L, float* __restrict__ out, int n) {
  int row = blockIdx.x * blockDim.x + threadIdx.x;
  if (row >= n) return;
  const float* l = L + (size_t)row * CP;
  float v[CDIM];
  float mx = -INFINITY;
  #pragma unroll
  for (int c = 0; c < CDIM; ++c) { v[c] = l[c]; mx = fmaxf(mx, v[c]); }
  float s = 0.0f;
  #pragma unroll
  for (int c = 0; c < CDIM; ++c) s += expf(v[c] - mx);
  float lse = mx + logf(s);
  float* o = out + (size_t)row * CDIM;
  #pragma unroll
  for (int c = 0; c < CDIM; ++c) o[c] = v[c] - lse;
}

extern "C" void kernel_launch(void* const* d_in, const int* in_sizes, int n_in,
                              void* d_out, int out_size, void* d_ws, size_t ws_size,
                              hipStream_t stream) {
  (void)n_in; (void)out_size; (void)ws_size;
  const float* x   = (const float*)d_in[0];
  const int*   ei  = (const int*)d_in[1];
  const float* w1  = (const float*)d_in[2];
  const float* b1  = (const float*)d_in[3];
  const float* wp0 = (const float*)d_in[4];
  const float* bp0 = (const float*)d_in[5];
  const float* wp1 = (const float*)d_in[6];
  const float* bp1 = (const float*)d_in[7];
  const float* wp2 = (const float*)d_in[8];
  const float* bp2 = (const float*)d_in[9];
  const float* w2  = (const float*)d_in[10];
  const float* b2  = (const float*)d_in[11];

  const int N = in_sizes[0] / F_IN;
  const int E = in_sizes[1] / 2;
  const int* src = ei;
  const int* dst = ei + E;

  float* h      = (float*)d_ws;                  // N*64
  float* h1     = h      + (size_t)N * HDIM;     // N*64
  float* h2     = h1     + (size_t)N * HDIM;     // N*64
  float* o192   = h2     + (size_t)N * HDIM;     // N*192
  float* logits = o192   + (size_t)N * K3;       // N*48
  float* deg    = logits + (size_t)N * CP;       // N
  float* dis    = deg    + N;                    // N
  float* norm   = dis    + N;                    // E
  float* w2p    = norm   + E;                    // 192*48
  float* b2p    = w2p    + K3 * CP;              // 48
  float* out    = (float*)d_out;

  const int gemmBlocks = (N + 31) / 32;

  // GCN normalization
  k_deg_init <<<(N + 255) / 256, 256, 0, stream>>>(deg, N);
  k_deg_count<<<(E + 255) / 256, 256, 0, stream>>>(dst, deg, E);
  k_dis      <<<(N + 255) / 256, 256, 0, stream>>>(deg, dis, N);
  k_norm     <<<(E + 255) / 256, 256, 0, stream>>>(src, dst, dis, norm, E);
  k_padw2    <<<(K3 * CP + CP + 255) / 256, 256, 0, stream>>>(w2, b2, w2p, b2p);

  // h = relu(x @ w1 + b1)
  k_lin1<<<gemmBlocks, 256, 0, stream>>>(x, w1, b1, h, N);

  // h1 = A_hat h ; h2 = A_hat h1
  {
    int initBlocks = (N * 16 + 255) / 256;
    unsigned long long et = (unsigned long long)E * 32ull;
    int edgeBlocks = (int)((et + 255ull) / 256ull);
    k_prop_init<<<initBlocks, 256, 0, stream>>>(h, dis, h1, N);
    k_prop_edge<<<edgeBlocks, 256, 0, stream>>>(src, dst, norm, h, h1, E);
    k_prop_init<<<initBlocks, 256, 0, stream>>>(h1, dis, h2, N);
    k_prop_edge<<<edgeBlocks, 256, 0, stream>>>(src, dst, norm, h1, h2, E);
  }

  // out192 = relu(concat(h@wp0+bp0, h1@wp1+bp1, h2@wp2+bp2))
  k_linp<<<gemmBlocks, 256, 0, stream>>>(h,  wp0, bp0, o192 +   0, N);
  k_linp<<<gemmBlocks, 256, 0, stream>>>(h1, wp1, bp1, o192 +  64, N);
  k_linp<<<gemmBlocks, 256, 0, stream>>>(h2, wp2, bp2, o192 + 128, N);

  // logits = out192 @ w2p + b2p ; log_softmax
  k_lin2<<<gemmBlocks, 192, 0, stream>>>(o192, w2p, b2p, logits, N);
  k_logsoftmax<<<(N + 255) / 256, 256, 0, stream>>>(logits, out, N);
}